// Seq2seqMASDLoss_24730421690616
// MI455X (gfx1250) — compile-verified
//
#include <hip/hip_runtime.h>
#include <hip/hip_bf16.h>
#include <cstdint>

namespace {
constexpr int   kB = 16, kS = 128, kV = 32000;
constexpr float kLog2e = 1.4426950408889634f;
constexpr float kLn2   = 0.6931471805599453f;
constexpr int   kRowChunk = 1024;                       // 256 threads * float4 per round
constexpr int   kRounds   = 31;                         // 31 * 1024 = 31744
constexpr int   kTailBase = kRounds * kRowChunk;        // 31744
constexpr int   kTailVec  = (kV - kTailBase) / 4;       // 64 float4 (256 floats)
}

// ---- CDNA5 async global->LDS (ASYNCcnt) ------------------------------------
__device__ __forceinline__ void async_ld128(unsigned lds_off, unsigned long long gaddr) {
  asm volatile("global_load_async_to_lds_b128 %0, %1, off"
               :: "v"(lds_off), "v"(gaddr)
               : "memory");
}
__device__ __forceinline__ void wait_async_le(int n) {
  if (n <= 0)      asm volatile("s_wait_asynccnt 0x0" ::: "memory");
  else if (n == 1) asm volatile("s_wait_asynccnt 0x1" ::: "memory");
  else             asm volatile("s_wait_asynccnt 0x2" ::: "memory");
}

// ---- per-thread online logsumexp + top-4 -----------------------------------
struct Red {
  float m, s;
  float v0, v1, v2, v3;
  int   i0, i1, i2, i3;
};

__device__ __forceinline__ void red_init(Red& r) {
  r.m = -__builtin_inff(); r.s = 0.f;
  r.v0 = r.v1 = r.v2 = r.v3 = -__builtin_inff();
  r.i0 = r.i1 = r.i2 = r.i3 = 0x7fffffff;
}

__device__ __forceinline__ void red_consume(Red& r, float x, int e) {
  if (x <= r.m) {
    r.s += exp2f((x - r.m) * kLog2e);
  } else {
    r.s = r.s * exp2f((r.m - x) * kLog2e) + 1.0f;
    r.m = x;
  }
  if (x > r.v3) {
    if (x > r.v0)      { r.v3=r.v2; r.i3=r.i2; r.v2=r.v1; r.i2=r.i1; r.v1=r.v0; r.i1=r.i0; r.v0=x; r.i0=e; }
    else if (x > r.v1) { r.v3=r.v2; r.i3=r.i2; r.v2=r.v1; r.i2=r.i1; r.v1=x; r.i1=e; }
    else if (x > r.v2) { r.v3=r.v2; r.i3=r.i2; r.v2=x; r.i2=e; }
    else               { r.v3=x; r.i3=e; }
  }
}

// ---- Phase 1: per-(b,s) row stats: normalizer + top-4 ----------------------
__global__ __launch_bounds__(256)
void rowstats_kernel(const float* __restrict__ logit, const int* __restrict__ tgt_lens,
                     float* __restrict__ rowc, float4* __restrict__ rowv,
                     int4* __restrict__ rowi) {
  const int row = blockIdx.x;                 // row = b*kS + s
  const int b = row >> 7, s = row & (kS - 1);
  int len = tgt_lens[b]; len = len < 2 ? 2 : (len > kS ? kS : len);
  if (s >= len - 1) return;                   // masked step: never read by phase 2

  const int tid = threadIdx.x;
  const float* rp = logit + (size_t)row * kV;

  __shared__ float4 stage[4][256];            // 16 KB quad-buffered async staging
  __shared__ float  rm[256], rs[256];
  __shared__ float  rv[256][4];
  __shared__ int    ri[256][4];

  Red rd; red_init(rd);

  // depth-2 async prefetch pipeline (quad buffer => no LDS reuse hazard)
  async_ld128((unsigned)(uintptr_t)&stage[0][tid],
              (unsigned long long)(rp + 0 * kRowChunk + tid * 4));
  async_ld128((unsigned)(uintptr_t)&stage[1][tid],
              (unsigned long long)(rp + 1 * kRowChunk + tid * 4));

  for (int r = 0; r < kRounds; ++r) {
    if (r + 2 < kRounds) {
      async_ld128((unsigned)(uintptr_t)&stage[(r + 2) & 3][tid],
                  (unsigned long long)(rp + (size_t)(r + 2) * kRowChunk + tid * 4));
    }
    int rem = kRounds - 1 - r;
    wait_async_le(rem < 2 ? rem : 2);         // round r complete for this wave's lanes
    float4 x = stage[r & 3][tid];             // ds_load_b128 (own lane's slice only)
    int e = r * kRowChunk + tid * 4;
    red_consume(rd, x.x, e);
    red_consume(rd, x.y, e + 1);
    red_consume(rd, x.z, e + 2);
    red_consume(rd, x.w, e + 3);
  }
  // tail: 31744..31999 via direct global loads (wave-uniform branch)
  if (tid < kTailVec) {
    const float4* tp = reinterpret_cast<const float4*>(rp + kTailBase);
    float4 x = tp[tid];
    int e = kTailBase + tid * 4;
    red_consume(rd, x.x, e);
    red_consume(rd, x.y, e + 1);
    red_consume(rd, x.z, e + 2);
    red_consume(rd, x.w, e + 3);
  }

  rm[tid] = rd.m; rs[tid] = rd.s;
  rv[tid][0] = rd.v0; rv[tid][1] = rd.v1; rv[tid][2] = rd.v2; rv[tid][3] = rd.v3;
  ri[tid][0] = rd.i0; ri[tid][1] = rd.i1; ri[tid][2] = rd.i2; ri[tid][3] = rd.i3;
  __syncthreads();

  // LDS tree reduction: merge (max, sumexp) and sorted top-4 lists
  for (int st = 128; st > 0; st >>= 1) {
    if (tid < st) {
      const int o = tid + st;
      float m1 = rm[tid], m2 = rm[o];
      float M  = fmaxf(m1, m2);
      rs[tid]  = rs[tid] * exp2f((m1 - M) * kLog2e) + rs[o] * exp2f((m2 - M) * kLog2e);
      rm[tid]  = M;
      float av[4], bw[4]; int ai[4], bi[4];
      for (int q = 0; q < 4; ++q) {
        av[q] = rv[tid][q]; ai[q] = ri[tid][q];
        bw[q] = rv[o][q];   bi[q] = ri[o][q];
      }
      float ov[4]; int oi[4];
      int ia = 0, ib = 0;
      for (int q = 0; q < 4; ++q) {          // 2-way merge, tie -> smaller index
        bool ta = (av[ia] > bw[ib]) || ((av[ia] == bw[ib]) && (ai[ia] < bi[ib]));
        if (ta) { ov[q] = av[ia]; oi[q] = ai[ia]; ++ia; }
        else    { ov[q] = bw[ib]; oi[q] = bi[ib]; ++ib; }
      }
      for (int q = 0; q < 4; ++q) { rv[tid][q] = ov[q]; ri[tid][q] = oi[q]; }
    }
    __syncthreads();
  }

  if (tid == 0) {
    rowc[row] = rm[0] + log2f(rs[0]) * kLn2;                 // max + ln(sumexp)
    rowv[row] = make_float4(rv[0][0], rv[0][1], rv[0][2], rv[0][3]);
    rowi[row] = make_int4(ri[0][0], ri[0][1], ri[0][2], ri[0][3]);
  }
}

// ---- Phase 2: per-batch beam recurrence + MASD loss (one wave32) -----------
__global__ __launch_bounds__(32)
void finalize_kernel(const float* __restrict__ rowc, const float4* __restrict__ rowv,
                     const int4* __restrict__ rowi, const float* __restrict__ asd,
                     const int* __restrict__ tgt_lens, float* __restrict__ out) {
  __shared__ float sh[kB];
  const int b = threadIdx.x;
  if (b < kB) {
    int len = tgt_lens[b]; len = len < 2 ? 2 : (len > kS ? kS : len);
    const float*  C  = rowc + b * kS;
    const float4* TV = rowv + b * kS;
    const int4*   TI = rowi + b * kS;

    float c0 = C[0]; float4 v0 = TV[0];
    float lq0 = v0.x - c0, lq1 = v0.y - c0, lq2 = v0.z - c0, lq3 = v0.w - c0;
    float a0 = lq0, a1 = lq1, a2 = lq2, a3 = lq3;   // per-rank gathered sums

    for (int s = 1; s < len - 1; ++s) {
      float c = C[s]; float4 tv = TV[s]; int4 ti = TI[s];
      float t[4]  = { tv.x - c, tv.y - c, tv.z - c, tv.w - c };
      int   vi[4] = { ti.x, ti.y, ti.z, ti.w };
      float lp[4] = { lq0, lq1, lq2, lq3 };

      float     bv[4] = { -__builtin_inff(), -__builtin_inff(), -__builtin_inff(), -__builtin_inff() };
      long long bf[4] = { 0x7fffffffffffffffLL, 0x7fffffffffffffffLL,
                          0x7fffffffffffffffLL, 0x7fffffffffffffffLL };
      float     bp[4] = { 0.f, 0.f, 0.f, 0.f };

      for (int k = 0; k < 4; ++k) {
        for (int j = 0; j < 4; ++j) {
          float     cv = lp[k] + t[j];
          long long cf = (long long)k * kV + vi[j];   // flattened beam*V index (tie-break)
          float     cp = t[j];                         // token's own log-prob
          if (cv > bv[3] || (cv == bv[3] && cf < bf[3])) {
            if (cv > bv[0] || (cv == bv[0] && cf < bf[0])) {
              bv[3]=bv[2]; bf[3]=bf[2]; bp[3]=bp[2];
              bv[2]=bv[1]; bf[2]=bf[1]; bp[2]=bp[1];
              bv[1]=bv[0]; bf[1]=bf[0]; bp[1]=bp[0];
              bv[0]=cv; bf[0]=cf; bp[0]=cp;
            } else if (cv > bv[1] || (cv == bv[1] && cf < bf[1])) {
              bv[3]=bv[2]; bf[3]=bf[2]; bp[3]=bp[2];
              bv[2]=bv[1]; bf[2]=bf[1]; bp[2]=bp[1];
              bv[1]=cv; bf[1]=cf; bp[1]=cp;
            } else if (cv > bv[2] || (cv == bv[2] && cf < bf[2])) {
              bv[3]=bv[2]; bf[3]=bf[2]; bp[3]=bp[2];
              bv[2]=cv; bf[2]=cf; bp[2]=cp;
            } else {
              bv[3]=cv; bf[3]=cf; bp[3]=cp;
            }
          }
        }
      }
      lq0 = bv[0]; lq1 = bv[1]; lq2 = bv[2]; lq3 = bv[3];
      a0 += bp[0]; a1 += bp[1]; a2 += bp[2]; a3 += bp[3];
    }

    // w = softmax over beam of topk_log_prob
    float mx = fmaxf(fmaxf(a0, a1), fmaxf(a2, a3));
    float e0 = exp2f((a0 - mx) * kLog2e), e1 = exp2f((a1 - mx) * kLog2e);
    float e2 = exp2f((a2 - mx) * kLog2e), e3 = exp2f((a3 - mx) * kLog2e);
    float inv = 1.0f / (e0 + e1 + e2 + e3);

    float s0 = asd[0 * kB + b], s1 = asd[1 * kB + b];
    float s2 = asd[2 * kB + b], s3 = asd[3 * kB + b];
    float mean = 0.25f * (s0 + s1 + s2 + s3);
    sh[b] = inv * (e0 * (s0 - mean) + e1 * (s1 - mean) +
                   e2 * (s2 - mean) + e3 * (s3 - mean));
  }
  __syncthreads();
  if (threadIdx.x == 0) {
    float tot = 0.f;
    for (int i = 0; i < kB; ++i) tot += sh[i];
    out[0] = tot * (1.0f / kB);
  }
}

extern "C" void kernel_launch(void* const* d_in, const int* in_sizes, int n_in,
                              void* d_out, int out_size, void* d_ws, size_t ws_size,
                              hipStream_t stream) {
  (void)in_sizes; (void)n_in; (void)out_size; (void)ws_size;
  const float* logit = (const float*)d_in[0];   // (16,128,32000) f32
  const float* asd   = (const float*)d_in[1];   // (4,16) f32
  const int*   tgt   = (const int*)d_in[2];     // (16,) i32
  // d_in[3] = beam_size (compile-time 4 here)

  char* ws = (char*)d_ws;
  float*  rowc = (float*)ws;                       // 2048 * 4  =  8192 B
  float4* rowv = (float4*)(ws + 8192);             // 2048 * 16 = 32768 B
  int4*   rowi = (int4*)(ws + 8192 + 32768);       // 2048 * 16 = 32768 B

  rowstats_kernel<<<kB * kS, 256, 0, stream>>>(logit, tgt, rowc, rowv, rowi);
  finalize_kernel<<<1, 32, 0, stream>>>(rowc, rowv, rowi, asd, tgt, (float*)d_out);
}